// RabitModel_eye_45380624450289
// MI455X (gfx1250) — compile-verified
//
#include <hip/hip_runtime.h>
#include <hip/hip_bf16.h>
#include <cstddef>

// ---------------- problem dims (from reference) ----------------
#define B 64
#define NVERT 40000
#define P 100
#define NJ 24
#define KP 128
#define COLS (3 * NVERT)   // 120000

typedef __attribute__((ext_vector_type(2))) float v2f;
typedef __attribute__((ext_vector_type(8))) float v8f;

// =====================================================================
// Kernel 1: v_posed[64, 120000] = beta[64,100] @ shapedirs[100,120000]
//           + v_template[120000]
// One wave per 16-column tile; 4 M-tiles (rows 0..63); K=100 in steps of 4
// via V_WMMA_F32_16X16X4_F32.
// =====================================================================
__global__ void shape_gemm_kernel(const float* __restrict__ beta,
                                  const float* __restrict__ shapedirs,
                                  const float* __restrict__ v_template,
                                  float* __restrict__ v_posed)
{
    const int lane = threadIdx.x & 31;
    const int wave = threadIdx.x >> 5;
    const int colTile = blockIdx.x * (blockDim.x >> 5) + wave;
    if (colTile >= COLS / 16) return;
    const int c0   = colTile * 16;
    const int half = lane >> 4;   // 0: K pair {k,k+1}, 1: K pair {k+2,k+3}
    const int l    = lane & 15;

    v8f acc[4];
    #pragma unroll
    for (int m = 0; m < 4; ++m) acc[m] = (v8f){0,0,0,0,0,0,0,0};

    for (int k = 0; k < P; k += 4) {
        const int kk = k + half * 2;
        // B fragment: rows kk, kk+1 of shapedirs at columns c0..c0+15
        v2f bf;
        bf.x = shapedirs[(size_t)(kk + 0) * COLS + c0 + l];
        bf.y = shapedirs[(size_t)(kk + 1) * COLS + c0 + l];
        #pragma unroll
        for (int m = 0; m < 4; ++m) {
            const int row = m * 16 + l;         // batch row 0..63
            v2f af;
            af.x = beta[row * P + kk + 0];
            af.y = beta[row * P + kk + 1];
            acc[m] = __builtin_amdgcn_wmma_f32_16x16x4_f32(
                false, af, false, bf, (short)0, acc[m], false, false);
        }
    }

    const float vt = v_template[c0 + l];
    #pragma unroll
    for (int m = 0; m < 4; ++m) {
        #pragma unroll
        for (int i = 0; i < 8; ++i) {
            const int row = m * 16 + i + half * 8;   // D layout: VGPR i -> M=i / M=i+8
            v_posed[(size_t)row * COLS + c0 + l] = acc[m][i] + vt;
        }
    }
}

// =====================================================================
// Kernel 2: rest joints J[64,24,3]: midpoint of min/max over 128 gathered
// verts of v_posed. grid = (B, 24); block = 128. Joint 0 = zeros.
// =====================================================================
__global__ void joints_kernel(const float* __restrict__ v_posed,
                              const int* __restrict__ joint_idx,
                              float* __restrict__ J)
{
    const int b = blockIdx.x;
    const int j = blockIdx.y;
    const int t = threadIdx.x;
    if (j == 0) {
        if (t < 3) J[(b * NJ + 0) * 3 + t] = 0.0f;
        return;
    }
    __shared__ float smn[3][KP];
    __shared__ float smx[3][KP];
    const int idx = joint_idx[(j - 1) * KP + t];
    #pragma unroll
    for (int c = 0; c < 3; ++c) {
        const float v = v_posed[(size_t)b * COLS + (size_t)idx * 3 + c];
        smn[c][t] = v;
        smx[c][t] = v;
    }
    __syncthreads();
    for (int s = KP / 2; s > 0; s >>= 1) {
        if (t < s) {
            #pragma unroll
            for (int c = 0; c < 3; ++c) {
                smn[c][t] = fminf(smn[c][t], smn[c][t + s]);
                smx[c][t] = fmaxf(smx[c][t], smx[c][t + s]);
            }
        }
        __syncthreads();
    }
    if (t < 3) J[(b * NJ + j) * 3 + t] = 0.5f * (smn[t][0] + smx[t][0]);
}

// =====================================================================
// Kernel 3: Rodrigues + kinematic chain + translation correction.
// One wave32 per batch; lane = joint. Parent tree (i-1)/2 has depth 4,
// so 4 level-synchronous compose steps using __shfl.
// Output G1[64,24,12] (top 3 rows of each 4x4).
// =====================================================================
__global__ void chain_kernel(const float* __restrict__ pose,     // [64,72]
                             const int* __restrict__ reorder,    // [24]
                             const int* __restrict__ parent,     // [24]
                             const float* __restrict__ J,        // [64,24,3]
                             float* __restrict__ G1)             // [64,24,12]
{
    const int b  = blockIdx.x;
    const int j  = threadIdx.x;           // 0..31
    const int jc = (j < NJ) ? j : (NJ - 1);
    const int jj = reorder[jc];

    const float r0 = pose[b * (NJ * 3) + jj * 3 + 0];
    const float r1 = pose[b * (NJ * 3) + jj * 3 + 1];
    const float r2 = pose[b * (NJ * 3) + jj * 3 + 2];
    const float th = fmaxf(sqrtf(r0 * r0 + r1 * r1 + r2 * r2), 1e-6f);
    const float x = r0 / th, y = r1 / th, z = r2 / th;
    const float cth = cosf(th), sth = sinf(th), C = 1.0f - cth;

    const float jx = J[(b * NJ + jc) * 3 + 0];
    const float jy = J[(b * NJ + jc) * 3 + 1];
    const float jz = J[(b * NJ + jc) * 3 + 2];
    const int   p  = parent[jc];
    const float tx = (jc == 0) ? jx : jx - J[(b * NJ + p) * 3 + 0];
    const float ty = (jc == 0) ? jy : jy - J[(b * NJ + p) * 3 + 1];
    const float tz = (jc == 0) ? jz : jz - J[(b * NJ + p) * 3 + 2];

    float G[12];
    G[0] = cth + C * x * x; G[1] = C * x * y - sth * z; G[2]  = C * x * z + sth * y; G[3]  = tx;
    G[4] = C * x * y + sth * z; G[5] = cth + C * y * y; G[6]  = C * y * z - sth * x; G[7]  = ty;
    G[8] = C * x * z - sth * y; G[9] = C * y * z + sth * x; G[10] = cth + C * z * z; G[11] = tz;

    const int mylevel = 31 - __clz(jc + 1);   // depth in binary tree
    for (int L = 1; L <= 4; ++L) {
        float Pm[12];
        #pragma unroll
        for (int q = 0; q < 12; ++q) Pm[q] = __shfl(G[q], p, 32);
        if (j < NJ && mylevel == L) {
            float Gn[12];
            #pragma unroll
            for (int r = 0; r < 3; ++r) {
                #pragma unroll
                for (int c = 0; c < 4; ++c) {
                    float acc = Pm[r * 4 + 0] * G[0 * 4 + c]
                              + Pm[r * 4 + 1] * G[1 * 4 + c]
                              + Pm[r * 4 + 2] * G[2 * 4 + c];
                    if (c == 3) acc += Pm[r * 4 + 3];
                    Gn[r * 4 + c] = acc;
                }
            }
            #pragma unroll
            for (int q = 0; q < 12; ++q) G[q] = Gn[q];
        }
    }

    if (j < NJ) {
        const float cx = G[0] * jx + G[1] * jy + G[2]  * jz;
        const float cy = G[4] * jx + G[5] * jy + G[6]  * jz;
        const float cz = G[8] * jx + G[9] * jy + G[10] * jz;
        G[3] -= cx; G[7] -= cy; G[11] -= cz;
        #pragma unroll
        for (int q = 0; q < 12; ++q) G1[((size_t)b * NJ + j) * 12 + q] = G[q];
    }
}

// =====================================================================
// Kernel 4: skinning. Per wave: 32 vertices of one batch.
// T[32,12] = weights[32,24] @ G1[b][24,12] via 12x V_WMMA_F32_16X16X4_F32
// (2 M-tiles x 6 K-steps), staged through LDS, then per-lane 3x4 transform.
// =====================================================================
__global__ void skin_kernel(const float* __restrict__ weights,  // [40000,24]
                            const float* __restrict__ G1,       // [64,24,12]
                            const float* __restrict__ v_posed,  // [64,120000]
                            const float* __restrict__ trans,    // [64,3]
                            float* __restrict__ posed)          // [64,40000,3]
{
    __shared__ float lds[8][32][17];
    const int wave = threadIdx.x >> 5;
    const int lane = threadIdx.x & 31;
    const int gw   = blockIdx.x * (blockDim.x >> 5) + wave;
    const int tilesPerBatch = NVERT / 32;              // 1250
    if (gw >= B * tilesPerBatch) return;
    const int b     = gw / tilesPerBatch;
    const int vtile = (gw % tilesPerBatch) * 32;
    const int half  = lane >> 4;
    const int l     = lane & 15;

    v8f acc0 = (v8f){0,0,0,0,0,0,0,0};
    v8f acc1 = (v8f){0,0,0,0,0,0,0,0};
    const float* g = G1 + (size_t)b * NJ * 12;

    for (int k = 0; k < NJ; k += 4) {
        const int kk = k + half * 2;
        v2f bf;
        bf.x = (l < 12) ? g[(kk + 0) * 12 + l] : 0.0f;
        bf.y = (l < 12) ? g[(kk + 1) * 12 + l] : 0.0f;
        v2f a0, a1;
        a0.x = weights[(size_t)(vtile + l) * NJ + kk + 0];
        a0.y = weights[(size_t)(vtile + l) * NJ + kk + 1];
        a1.x = weights[(size_t)(vtile + 16 + l) * NJ + kk + 0];
        a1.y = weights[(size_t)(vtile + 16 + l) * NJ + kk + 1];
        acc0 = __builtin_amdgcn_wmma_f32_16x16x4_f32(
            false, a0, false, bf, (short)0, acc0, false, false);
        acc1 = __builtin_amdgcn_wmma_f32_16x16x4_f32(
            false, a1, false, bf, (short)0, acc1, false, false);
    }

    #pragma unroll
    for (int i = 0; i < 8; ++i) {
        lds[wave][i + half * 8][l]      = acc0[i];   // vertices 0..15 of tile
        lds[wave][16 + i + half * 8][l] = acc1[i];   // vertices 16..31
    }
    __syncthreads();

    const int v = vtile + lane;
    float T[12];
    #pragma unroll
    for (int c = 0; c < 12; ++c) T[c] = lds[wave][lane][c];

    const float vx = v_posed[(size_t)b * COLS + (size_t)v * 3 + 0];
    const float vy = v_posed[(size_t)b * COLS + (size_t)v * 3 + 1];
    const float vz = v_posed[(size_t)b * COLS + (size_t)v * 3 + 2];

    #pragma unroll
    for (int c = 0; c < 3; ++c) {
        const float o = T[c * 4 + 0] * vx + T[c * 4 + 1] * vy +
                        T[c * 4 + 2] * vz + T[c * 4 + 3] + trans[b * 3 + c];
        posed[((size_t)b * NVERT + v) * 3 + c] = o;
    }
}

// =====================================================================
// Kernel 5: skeleton[64,31,3]: row 0 zeros, rows 1..23 keypoints(joint_idx),
// rows 24..30 keypoints(add_idx), computed on posed output.
// =====================================================================
__global__ void skeleton_kernel(const float* __restrict__ posed,
                                const int* __restrict__ joint_idx,
                                const int* __restrict__ add_idx,
                                float* __restrict__ skel)        // [64,31,3]
{
    const int b = blockIdx.x;
    const int j = blockIdx.y;    // 0..30
    const int t = threadIdx.x;
    if (j == 0) {
        if (t < 3) skel[(b * 31 + 0) * 3 + t] = 0.0f;
        return;
    }
    __shared__ float smn[3][KP];
    __shared__ float smx[3][KP];
    const int idx = (j <= 23) ? joint_idx[(j - 1) * KP + t]
                              : add_idx[(j - 24) * KP + t];
    #pragma unroll
    for (int c = 0; c < 3; ++c) {
        const float v = posed[((size_t)b * NVERT + idx) * 3 + c];
        smn[c][t] = v;
        smx[c][t] = v;
    }
    __syncthreads();
    for (int s = KP / 2; s > 0; s >>= 1) {
        if (t < s) {
            #pragma unroll
            for (int c = 0; c < 3; ++c) {
                smn[c][t] = fminf(smn[c][t], smn[c][t + s]);
                smx[c][t] = fmaxf(smx[c][t], smx[c][t + s]);
            }
        }
        __syncthreads();
    }
    if (t < 3) skel[(b * 31 + j) * 3 + t] = 0.5f * (smn[t][0] + smx[t][0]);
}

// =====================================================================
extern "C" void kernel_launch(void* const* d_in, const int* in_sizes, int n_in,
                              void* d_out, int out_size, void* d_ws, size_t ws_size,
                              hipStream_t stream)
{
    const float* beta       = (const float*)d_in[0];  // [64,100]
    const float* pose       = (const float*)d_in[1];  // [64,72]
    const float* trans      = (const float*)d_in[2];  // [64,3]
    const float* shapedirs  = (const float*)d_in[3];  // [100,120000]
    const float* v_template = (const float*)d_in[4];  // [120000]
    const float* weights    = (const float*)d_in[5];  // [40000,24]
    const int*   reorder    = (const int*)d_in[6];    // [24]
    const int*   parent     = (const int*)d_in[7];    // [24]
    const int*   joint_idx  = (const int*)d_in[8];    // [23,128]
    const int*   add_idx    = (const int*)d_in[9];    // [7,128]

    float* out_posed = (float*)d_out;                      // 64*40000*3
    float* out_skel  = out_posed + (size_t)B * NVERT * 3;  // 64*31*3

    float* ws      = (float*)d_ws;
    float* v_posed = ws;                                   // 64*120000
    float* J       = v_posed + (size_t)B * COLS;           // 64*24*3
    float* G1      = J + (size_t)B * NJ * 3;               // 64*24*12

    // 1) shape blend GEMM (WMMA fp32)
    {
        const int tiles = COLS / 16;          // 7500 waves
        const int waves_per_block = 8;        // 256 threads
        const int blocks = (tiles + waves_per_block - 1) / waves_per_block;
        shape_gemm_kernel<<<blocks, 256, 0, stream>>>(beta, shapedirs, v_template, v_posed);
    }
    // 2) rest joints
    joints_kernel<<<dim3(B, NJ), KP, 0, stream>>>(v_posed, joint_idx, J);
    // 3) rodrigues + chain
    chain_kernel<<<B, 32, 0, stream>>>(pose, reorder, parent, J, G1);
    // 4) skinning (WMMA fp32)
    {
        const int waves = B * (NVERT / 32);   // 80000
        const int blocks = waves / 8;         // 10000 blocks of 256 threads
        skin_kernel<<<blocks, 256, 0, stream>>>(weights, G1, v_posed, trans, out_posed);
    }
    // 5) skeleton keypoints
    skeleton_kernel<<<dim3(B, 31), KP, 0, stream>>>(out_posed, joint_idx, add_idx, out_skel);
}